// Operator_14035953123928
// MI455X (gfx1250) — compile-verified
//
#include <hip/hip_runtime.h>

typedef float v2f __attribute__((ext_vector_type(2)));
typedef float v8f __attribute__((ext_vector_type(8)));

#define Bsz 8
#define Psz 32
#define Tsz 24
#define Csz 4
#define Xsz 2048
#define ROW (Csz * Xsz)        /* 8192 floats per t-row            */
#define GRAM (Tsz * Tsz)       /* 576                               */
#define NBP (Bsz * Psz)        /* 256 (B,p) slabs                   */
#define WOFF (NBP * GRAM)      /* float offset of w-vectors in ws   */

// ---------------------------------------------------------------------------
// Kernel 1: per-(B,p) Gram partials  G = X2 * X2^T  via V_WMMA_F32_16X16X4_F32
// 256 blocks x 256 threads (8 waves). Wave w reduces columns [w*1024, w*1024+1024).
// 24 rows padded to 32 -> 2x2 tiles of 16x16; G10 recovered by symmetry.
// Padding rows 24..31 read the mu row so x2 = 0 with no exec-mask predication.
// ---------------------------------------------------------------------------
__global__ __launch_bounds__(256) void gram_wmma_kernel(
    const float* __restrict__ x, float* __restrict__ ws)
{
    __shared__ float g[8][24][25];

    const int lane = threadIdx.x & 31;
    const int w    = threadIdx.x >> 5;
    const int m    = lane & 15;      // row within 16-row tile / column index
    const int hs   = lane >> 4;      // half-wave: 0 -> K=0,1 ; 1 -> K=2,3

    const float* base = x + (size_t)blockIdx.x * (Tsz * ROW);
    const float* r0   = base + (size_t)m * ROW;                    // rows 0..15
    const int    row1 = (m < 8) ? (16 + m) : 23;                   // clamp pad->mu
    const float* r1   = base + (size_t)row1 * ROW;                 // rows 16..23
    const float* rmu  = base + (size_t)23 * ROW;                   // mu row

    v8f a00 = {0.f,0.f,0.f,0.f,0.f,0.f,0.f,0.f};
    v8f a01 = {0.f,0.f,0.f,0.f,0.f,0.f,0.f,0.f};
    v8f a11 = {0.f,0.f,0.f,0.f,0.f,0.f,0.f,0.f};

    const int k0 = w * 1024;
    #pragma unroll 4
    for (int s = 0; s < 256; ++s) {
        const int kb = k0 + s * 4 + 2 * hs;
        v2f muv = *(const v2f*)(rmu + kb);
        v2f f0  = *(const v2f*)(r0  + kb);
        v2f f1  = *(const v2f*)(r1  + kb);
        v2f t0  = f0 - muv;                 // x2 rows 0..15
        v2f t1  = f1 - muv;                 // x2 rows 16..23; pad rows -> 0
        // For G = A*A^T the 16x4 A-fragment and 4x16 B-fragment layouts are
        // lane-identical, so t0/t1 feed both operands directly.
        a00 = __builtin_amdgcn_wmma_f32_16x16x4_f32(false, t0, false, t0,
                                                    (short)0, a00, false, false);
        a01 = __builtin_amdgcn_wmma_f32_16x16x4_f32(false, t0, false, t1,
                                                    (short)0, a01, false, false);
        a11 = __builtin_amdgcn_wmma_f32_16x16x4_f32(false, t1, false, t1,
                                                    (short)0, a11, false, false);
    }

    // C/D layout: VGPR r -> M=r (lanes 0..15), M=r+8 (lanes 16..31); N = lane%16.
    const int rbase = 8 * hs;
    #pragma unroll
    for (int r = 0; r < 8; ++r) {
        g[w][rbase + r][m] = a00[r];                       // rows 0..15, cols 0..15
        if (m < 8) g[w][rbase + r][16 + m] = a01[r];       // rows 0..15, cols 16..23
    }
    if (hs == 0 && m < 8) {
        #pragma unroll
        for (int r = 0; r < 8; ++r) g[w][16 + r][16 + m] = a11[r]; // rows/cols 16..23
    }
    __syncthreads();

    // Deterministic cross-wave reduction; G[t][u] for t>=16,u<16 via symmetry.
    for (int e = threadIdx.x; e < GRAM; e += 256) {
        const int t = e / 24, u = e % 24;
        float ssum = 0.f;
        #pragma unroll
        for (int ww = 0; ww < 8; ++ww)
            ssum += (t >= 16 && u < 16) ? g[ww][u][t] : g[ww][t][u];
        ws[(size_t)blockIdx.x * GRAM + e] = ssum;
    }
}

// ---------------------------------------------------------------------------
// Kernel 2: per-batch 24x24 eigensolve (cyclic Jacobi) + small-matrix algebra.
// 8 blocks x 32 threads (one wave per batch; wave-synchronous LDS).
// Produces w[B][24] such that out = sum_t w[t]*(x[t]-mu) + mu.
// ---------------------------------------------------------------------------
__global__ __launch_bounds__(32) void solve_kernel(
    const float* __restrict__ ws_in,
    const float* __restrict__ Kd, const float* __restrict__ Vd,
    const float* __restrict__ Qd, const float* __restrict__ Dd,
    float* __restrict__ wout)
{
    __shared__ float A[24][26], C0[24][26], V[24][26], VT[24][26], X3[24][26];
    __shared__ float kk[24], vv[24], qv[24], dm[24], x4[24], sv[24], dd[24];
    __shared__ int   perm[24];

    const int lane = threadIdx.x;
    const int b    = blockIdx.x;

    // 1) reduce the 32 per-p Gram partials -> vssv (kept twice: work + original)
    for (int e = lane; e < GRAM; e += 32) {
        const int t = e / 24, u = e % 24;
        float ssum = 0.f;
        for (int p = 0; p < 32; ++p)
            ssum += ws_in[(size_t)(b * 32 + p) * GRAM + e];
        A[t][u] = ssum; C0[t][u] = ssum;
    }
    if (lane < 24)
        for (int t = 0; t < 24; ++t) V[t][lane] = (t == lane) ? 1.f : 0.f;
    __syncthreads();

    // 2) cyclic Jacobi sweeps (branchless rotation; single wave => lockstep)
    for (int sweep = 0; sweep < 10; ++sweep) {
        for (int p = 0; p < 23; ++p) {
            for (int q = p + 1; q < 24; ++q) {
                const float apq = A[p][q], app = A[p][p], aqq = A[q][q];
                float c = 1.f, s = 0.f;
                if (fabsf(apq) > 1e-20f) {
                    const float tau = (aqq - app) / (2.f * apq);
                    const float tt  = ((tau >= 0.f) ? 1.f : -1.f) /
                                      (fabsf(tau) + sqrtf(1.f + tau * tau));
                    c = 1.f / sqrtf(1.f + tt * tt);
                    s = tt * c;
                }
                if (lane < 24) {                      // rows p,q
                    const float xp = A[p][lane], xq = A[q][lane];
                    A[p][lane] = c * xp - s * xq;
                    A[q][lane] = s * xp + c * xq;
                }
                __syncthreads();
                if (lane < 24) {                      // cols p,q and V <- V*J
                    const float xp = A[lane][p], xq = A[lane][q];
                    A[lane][p] = c * xp - s * xq;
                    A[lane][q] = s * xp + c * xq;
                    const float vp = V[lane][p], vq = V[lane][q];
                    V[lane][p] = c * vp - s * vq;
                    V[lane][q] = s * vp + c * vq;
                }
                __syncthreads();
            }
        }
    }

    if (lane < 24) dd[lane] = A[lane][lane];
    __syncthreads();

    // 3) sort eigenpairs descending; canonical sign; s = sqrt(max(e,1))
    if (lane < 24) {
        int rank = 0;
        for (int i = 0; i < 24; ++i) {
            const float di = dd[i];
            if (di > dd[lane] || (di == dd[lane] && i < lane)) ++rank;
        }
        perm[rank] = lane;
    }
    __syncthreads();
    if (lane < 24) {
        const int src = perm[lane];
        float best = 0.f, sgn = 1.f;
        for (int t = 0; t < 24; ++t) {
            const float vtl = V[t][src];
            if (fabsf(vtl) > best) { best = fabsf(vtl); sgn = (vtl >= 0.f) ? 1.f : -1.f; }
        }
        for (int t = 0; t < 24; ++t) VT[lane][t] = sgn * V[t][src];
        float e = dd[src]; if (e < 1.f) e = 1.f;
        sv[lane] = sqrtf(e);
    }
    __syncthreads();

    // 4) x3 = diag(1/s) * vt * vssv
    for (int e = lane; e < GRAM; e += 32) {
        const int a = e / 24, u = e % 24;
        float ssum = 0.f;
        for (int t = 0; t < 24; ++t) ssum += VT[a][t] * C0[t][u];
        X3[a][u] = ssum / sv[a];
    }
    __syncthreads();

    // 5) k,v,q,dmd contractions: r[m] = sum_{a,u} R[a][u][m] * x3[a][u]
    if (lane < 24) {
        float sk = 0.f, svv = 0.f, sq = 0.f, sd = 0.f;
        for (int a = 0; a < 24; ++a)
            for (int u = 0; u < 24; ++u) {
                const float x3v = X3[a][u];
                const size_t off = ((size_t)(a * 24 + u)) * 24 + lane;
                sk  += Kd[off] * x3v;
                svv += Vd[off] * x3v;
                sq  += Qd[off] * x3v;
                sd  += Dd[off] * x3v;
            }
        kk[lane] = sk; vv[lane] = svv; qv[lane] = sq; dm[lane] = sd;
    }
    __syncthreads();

    // 6) x4 = (q.k) * v + dmd_term
    float dot = 0.f;
    for (int i = 0; i < 24; ++i) dot += qv[i] * kk[i];
    if (lane < 24) x4[lane] = dot * vv[lane] + dm[lane];
    __syncthreads();

    // 7) w[t] = sum_m x4[m] * vt[m][t] / s[m]
    if (lane < 24) {
        float ssum = 0.f;
        for (int mm = 0; mm < 24; ++mm) ssum += (x4[mm] / sv[mm]) * VT[mm][lane];
        wout[b * 24 + lane] = ssum;
    }
}

// ---------------------------------------------------------------------------
// Kernel 3: out[b,p,0,c] = sum_t w[b,t]*(x[b,p,t,c,0]-mu) + mu,  mu = x[b,p,23,c,0]
// ---------------------------------------------------------------------------
__global__ __launch_bounds__(256) void out_kernel(
    const float* __restrict__ x, const float* __restrict__ wv,
    float* __restrict__ out)
{
    const int idx = blockIdx.x * 256 + threadIdx.x;
    if (idx >= Bsz * Psz * Csz) return;
    const int b = idx >> 7;
    const int p = (idx >> 2) & 31;
    const int c = idx & 3;
    const float* basep = x + ((size_t)(b * 32 + p) * Tsz) * ROW + (size_t)c * Xsz;
    const float mu = basep[(size_t)23 * ROW];
    float acc = 0.f;
    #pragma unroll
    for (int t = 0; t < 24; ++t)
        acc += wv[b * 24 + t] * (basep[(size_t)t * ROW] - mu);
    out[idx] = acc + mu;
}

// ---------------------------------------------------------------------------
extern "C" void kernel_launch(void* const* d_in, const int* in_sizes, int n_in,
                              void* d_out, int out_size, void* d_ws, size_t ws_size,
                              hipStream_t stream)
{
    const float* x  = (const float*)d_in[0];
    const float* Kd = (const float*)d_in[1];
    const float* Vd = (const float*)d_in[2];
    const float* Qd = (const float*)d_in[3];
    const float* Dd = (const float*)d_in[4];
    float* wsf  = (float*)d_ws;
    float* wvec = wsf + WOFF;

    gram_wmma_kernel<<<NBP, 256, 0, stream>>>(x, wsf);
    solve_kernel<<<Bsz, 32, 0, stream>>>(wsf, Kd, Vd, Qd, Dd, wvec);
    out_kernel<<<(Bsz * Psz * Csz + 255) / 256, 256, 0, stream>>>(x, wvec, (float*)d_out);
}